// Torch_Ops_Aten_Topk_ValuesModule_66236985639522
// MI455X (gfx1250) — compile-verified
//
#include <hip/hip_runtime.h>
#include <hip/hip_bf16.h>
#include <stdint.h>

// Row-wise top-k (k=50) over V=50257 fp32, rows = 2*4096.
// Single streaming pass per row (HBM-bound, ~71us roofline at 23.3 TB/s).
// Non-temporal b128 stream -> threshold filter -> LDS candidate buffer
// (LDS atomics) -> histogram radix-select compaction -> tiny bitonic sort.
// wave32: 256-thread block = 8 waves; ~17KB LDS/block << 320KB/WGP.

namespace {

typedef float v4f __attribute__((ext_vector_type(4)));  // NT-loadable 16B vector

constexpr int kV    = 50257;      // vocab (reduced axis)
constexpr int kRows = 2 * 4096;   // B * S
constexpr int kK    = 50;
constexpr int BLOCK = 256;        // 8 x wave32
constexpr int CAP   = 2048;       // candidate key buffer (uint64)
constexpr int CHUNK = 4 * BLOCK;  // elements consumed per streamed chunk
constexpr int PERT  = CAP / BLOCK;

// Monotone float -> u32: larger float => larger unsigned.
__device__ __forceinline__ unsigned mono_u32(float f) {
  unsigned b = __float_as_uint(f);
  return (b & 0x80000000u) ? ~b : (b | 0x80000000u);
}
__device__ __forceinline__ float inv_mono(unsigned u) {
  unsigned b = (u & 0x80000000u) ? (u ^ 0x80000000u) : ~u;
  return __uint_as_float(b);
}

// Descending bitonic sort of keys[0..N), N power of two, block-uniform.
__device__ __forceinline__ void bitonic_desc(uint64_t* keys, int N, int tid) {
  for (int k2 = 2; k2 <= N; k2 <<= 1) {
    for (int j = k2 >> 1; j > 0; j >>= 1) {
      __syncthreads();
      for (int i = tid; i < N; i += BLOCK) {
        int ixj = i ^ j;
        if (ixj > i) {
          uint64_t a = keys[i], b = keys[ixj];
          bool up = ((i & k2) == 0);
          if (up ? (a < b) : (a > b)) { keys[i] = b; keys[ixj] = a; }
        }
      }
    }
  }
  __syncthreads();
}

// 2-level (8+8 bit) histogram radix-select of the K-th largest key's top 16
// bits, then compact survivors (top16 >= threshold) to the buffer front.
// Sets *p_cnt = #survivors (>= K), *p_thresh = 32-bit streaming threshold.
__device__ __forceinline__ void select_compact(uint64_t* keys, unsigned* hist,
                                               int* p_cnt, unsigned* p_thresh,
                                               int tid) {
  __shared__ int sh_b1, sh_rem;
  __shared__ unsigned sh_thr16;

  int cnt = *p_cnt; if (cnt > CAP) cnt = CAP;

  uint64_t mine[PERT];
  #pragma unroll
  for (int s = 0; s < PERT; ++s) {
    int p = tid + s * BLOCK;
    mine[s] = (p < cnt) ? keys[p] : 0ull;
  }

  // level 1: key bits [63:56]
  for (int b = tid; b < 256; b += BLOCK) hist[b] = 0u;
  __syncthreads();
  #pragma unroll
  for (int s = 0; s < PERT; ++s) {
    int p = tid + s * BLOCK;
    if (p < cnt) atomicAdd(&hist[(unsigned)(mine[s] >> 56)], 1u);
  }
  __syncthreads();
  if (tid == 0) {
    int acc = 0, b1 = 0, rem = kK;
    for (int b = 255; b >= 0; --b) {
      int c = (int)hist[b];
      if (acc + c >= kK) { b1 = b; rem = kK - acc; break; }
      acc += c;
    }
    sh_b1 = b1; sh_rem = rem;
  }
  __syncthreads();
  const int b1 = sh_b1, rem = sh_rem;

  // level 2: key bits [55:48] within the boundary bin
  for (int b = tid; b < 256; b += BLOCK) hist[b] = 0u;
  __syncthreads();
  #pragma unroll
  for (int s = 0; s < PERT; ++s) {
    int p = tid + s * BLOCK;
    if (p < cnt && (unsigned)(mine[s] >> 56) == (unsigned)b1)
      atomicAdd(&hist[(unsigned)(mine[s] >> 48) & 0xFFu], 1u);
  }
  __syncthreads();
  if (tid == 0) {
    int acc = 0, b2 = 0;
    for (int b = 255; b >= 0; --b) {
      acc += (int)hist[b];
      if (acc >= rem) { b2 = b; break; }
    }
    sh_thr16 = ((unsigned)b1 << 8) | (unsigned)b2;
    *p_cnt = 0;
  }
  __syncthreads();
  const unsigned thr16 = sh_thr16;

  // compact survivors back to the front (safe: sources cached in registers)
  uint64_t keep[PERT]; int m = 0;
  #pragma unroll
  for (int s = 0; s < PERT; ++s) {
    int p = tid + s * BLOCK;
    if (p < cnt && (unsigned)(mine[s] >> 48) >= thr16) keep[m++] = mine[s];
  }
  int base = 0;
  if (m) base = atomicAdd(p_cnt, m);
  for (int c = 0; c < m; ++c) keys[base + c] = keep[c];
  if (tid == 0) *p_thresh = thr16 << 16;
  __syncthreads();
}

__global__ __launch_bounds__(BLOCK)
void topk_rows(const float* __restrict__ x,
               float* __restrict__ out_vals,
               long long* __restrict__ out_idx) {
  __shared__ uint64_t s_keys[CAP];   // 16 KB
  __shared__ unsigned s_hist[256];   // 1 KB
  __shared__ int s_cnt;
  __shared__ unsigned s_thresh;

  const int tid = threadIdx.x;
  const int row = blockIdx.x;
  const float* __restrict__ xr = x + (size_t)row * kV;

  if (tid == 0) { s_cnt = 0; s_thresh = 0u; }
  __syncthreads();

  // Alignment peel so the streamed body uses 16B-aligned b128 loads.
  const int mis  = (int)(((uintptr_t)xr >> 2) & 3u);
  int head = (4 - mis) & 3; if (head > kV) head = kV;
  const int nvec = (kV - head) >> 2;
  const int tail = head + 4 * nvec;

  for (int e = tid; e < head; e += BLOCK) {
    unsigned u = mono_u32(__builtin_nontemporal_load(xr + e));
    int p = atomicAdd(&s_cnt, 1);
    s_keys[p] = ((uint64_t)u << 32) | (uint64_t)(unsigned)~(unsigned)e;
  }
  for (int e = tail + tid; e < kV; e += BLOCK) {
    unsigned u = mono_u32(__builtin_nontemporal_load(xr + e));
    int p = atomicAdd(&s_cnt, 1);
    s_keys[p] = ((uint64_t)u << 32) | (uint64_t)(unsigned)~(unsigned)e;
  }

  const v4f* __restrict__ xv = (const v4f*)(xr + head);
  for (int base = 0; base < nvec; base += BLOCK) {
    // One barrier per chunk: check overflow with the count as of this
    // barrier. Invariant: cnt <= CAP-CHUNK after this block, appends add
    // <= CHUNK, so cnt <= CAP always.
    __syncthreads();
    if (s_cnt > CAP - CHUNK) {               // block-uniform
      select_compact(s_keys, s_hist, &s_cnt, &s_thresh, tid);
      if (s_cnt > CAP - CHUNK) {
        // pathological tie fallback: exact top-K via full sort
        int cnt = s_cnt; if (cnt > CAP) cnt = CAP;
        for (int p2 = cnt + tid; p2 < CAP; p2 += BLOCK) s_keys[p2] = 0ull;
        bitonic_desc(s_keys, CAP, tid);
        if (tid == 0) { s_cnt = kK; s_thresh = (unsigned)(s_keys[kK - 1] >> 32); }
        __syncthreads();
      }
    }
    const unsigned thr = s_thresh;   // stable: only compaction writes it
    const int i = base + tid;
    if (i < nvec) {
      if (i + 2 * BLOCK < nvec)
        __builtin_prefetch((const void*)(xv + i + 2 * BLOCK), 0, 1);
      const v4f v = __builtin_nontemporal_load(xv + i);  // NT b128 stream
      const float f[4] = {v.x, v.y, v.z, v.w};
      const int e0 = head + 4 * i;
      uint64_t loc[4]; int m = 0;
      #pragma unroll
      for (int c = 0; c < 4; ++c) {
        const unsigned u = mono_u32(f[c]);
        if (u >= thr)
          loc[m++] = ((uint64_t)u << 32) |
                     (uint64_t)(unsigned)~(unsigned)(e0 + c);
      }
      if (m) {
        int p = atomicAdd(&s_cnt, m);
        #pragma unroll
        for (int c = 0; c < 4; ++c)
          if (c < m && p + c < CAP) s_keys[p + c] = loc[c];
      }
    }
  }
  __syncthreads();

  // Final: shrink to ~K survivors, then sort just a tiny power-of-two region.
  select_compact(s_keys, s_hist, &s_cnt, &s_thresh, tid);
  int m = s_cnt; if (m > CAP) m = CAP;
  int N = 64; while (N < m) N <<= 1;
  for (int p = m + tid; p < N; p += BLOCK) s_keys[p] = 0ull;
  bitonic_desc(s_keys, N, tid);

  if (tid < kK) {
    const uint64_t key = s_keys[tid];
    const unsigned u = (unsigned)(key >> 32);
    out_vals[(size_t)row * kK + tid] = inv_mono(u);
    out_idx[(size_t)row * kK + tid] =
        (long long)(unsigned)~(unsigned)(key & 0xFFFFFFFFull);
  }
}

} // namespace

extern "C" void kernel_launch(void* const* d_in, const int* in_sizes, int n_in,
                              void* d_out, int out_size, void* d_ws, size_t ws_size,
                              hipStream_t stream) {
  (void)in_sizes; (void)n_in; (void)d_ws; (void)ws_size; (void)out_size;
  const float* x = (const float*)d_in[0];
  float* out_vals = (float*)d_out;                                  // values fp32
  long long* out_idx = (long long*)(out_vals + (size_t)kRows * kK); // indices i64
  topk_rows<<<dim3(kRows), dim3(BLOCK), 0, stream>>>(x, out_vals, out_idx);
}